// GDR_88029649699073
// MI455X (gfx1250) — compile-verified
//
#include <hip/hip_runtime.h>
#include <cstdint>
#include <cstddef>

// ---------------------------------------------------------------------------
// GCN autoencoder for MI455X (gfx1250, wave32).
// GEMMs use V_WMMA_F32_16X16X4_F32 (exact fp32 semantics; the workload is
// bandwidth-bound: ~1.5 GB traffic / 23.3 TB/s ~= 70us, GEMM FLOPs ~8G are
// negligible, so fp32 matrix ops are the right precision). Edge scatter uses
// fp32 no-return atomics (global_atomic_add_f32) plus global_prefetch_b8 to
// hide the 1.3 GB gather. The skinny z-GEMM (K=20000) is split-K with atomic
// reduction so it doesn't serialize on 32 waves. Tile coordinates are forced
// into SGPRs via readfirstlane so the k-loop is a uniform s_cbranch loop
// (EXEC stays all-1s through every WMMA, as the ISA requires) and unrolls.
// ---------------------------------------------------------------------------

#define NGENES 20000
#define NBATCH 128
#define FMID   512      // MID*BATCH
#define TDIM   64
#define RMID   256      // MID*TDIM
#define NEDGES 640000

typedef float v2f __attribute__((ext_vector_type(2)));
typedef float v8f __attribute__((ext_vector_type(8)));

__device__ __forceinline__ float celu1(float v) {
  // jax.nn.celu with alpha=1
  return v > 0.0f ? v : (__expf(v) - 1.0f);
}

// ---------------------------------------------------------------------------
// Generic fp32 WMMA GEMM: C[M,N] (+)= op(A)[M,K] @ B[K,N]  (+bias, +celu)
// One wave computes one 16x16 tile slice; M,N multiples of 16, K/nsplit
// multiple of 4. A-fragment (16x4, ISA 7.12.2): lanes 0-15 hold M=lane
// {K=k0,k0+1}, lanes 16-31 hold M=lane-16 {K=k0+2,k0+3}; B (4x16) symmetric;
// D: VGPR r -> M = r (+8 upper half-wave), N = lane&15.
// ATOMIC_OUT: split-K partial tiles reduced with global_atomic_add_f32.
// ---------------------------------------------------------------------------
template <bool TRANS_A, bool CELU_A, bool CELU_OUT, bool HAS_BIAS, bool ATOMIC_OUT>
__global__ __launch_bounds__(256) void wmma_gemm_f32(
    const float* __restrict__ A, const float* __restrict__ B,
    const float* __restrict__ bias, float* __restrict__ C,
    int M, int N, int K, int lda, int ldb, int ldc, int nsplit) {
  const int wavesPerBlk = blockDim.x >> 5;
  // Force wave id scalar: all tile/split math lands in SGPRs, the k-loop is
  // wave-uniform (s_cmp/s_cbranch, no EXEC masking around v_wmma).
  const int wave = __builtin_amdgcn_readfirstlane(
      blockIdx.x * wavesPerBlk + ((int)threadIdx.x >> 5));
  const int tilesN = N >> 4;
  const int tiles  = (M >> 4) * tilesN;
  if (wave >= tiles * nsplit) return;   // uniform per wave: EXEC stays all-1s

  const int split = wave / tiles;
  const int tile  = wave - split * tiles;
  const int kc    = K / nsplit;         // multiple of 4 by construction
  const int kbeg  = split * kc;
  const int kend  = kbeg + kc;

  const int lane = (int)threadIdx.x & 31;
  const int half = lane >> 4;           // 0: lanes 0-15, 1: lanes 16-31
  const int l    = lane & 15;
  const int tm = tile / tilesN;
  const int tn = tile - tm * tilesN;
  const int m0 = tm << 4, n0 = tn << 4;
  const int mA = m0 + l;
  const int nB = n0 + l;

  v8f acc = {};
#pragma unroll 4
  for (int k0 = kbeg; k0 < kend; k0 += 4) {
    const int ka = k0 + (half << 1);
    v2f a, b;
    if (TRANS_A) {                      // A stored K-major: A(m,k)=A[k*lda+m]
      a.x = A[(size_t)ka * lda + mA];
      a.y = A[(size_t)(ka + 1) * lda + mA];
    } else {                            // A row-major, 8B-aligned pair -> b64
      a = *(const v2f*)(A + (size_t)mA * lda + ka);
    }
    if (CELU_A) { a.x = celu1(a.x); a.y = celu1(a.y); }
    b.x = B[(size_t)ka * ldb + nB];
    b.y = B[(size_t)(ka + 1) * ldb + nB];
    // (neg_a, A, neg_b, B, c_mod, C, reuse_a, reuse_b)
    acc = __builtin_amdgcn_wmma_f32_16x16x4_f32(
        false, a, false, b, (short)0, acc, false, false);
  }

  if constexpr (ATOMIC_OUT) {
#pragma unroll
    for (int r = 0; r < 8; ++r)
      unsafeAtomicAdd(&C[(size_t)(m0 + (half << 3) + r) * ldc + nB], acc[r]);
  } else {
    float bn;
    if constexpr (HAS_BIAS) bn = bias[nB];
#pragma unroll
    for (int r = 0; r < 8; ++r) {
      float v = acc[r];
      if constexpr (HAS_BIAS) v += bn;
      if constexpr (CELU_OUT) v = celu1(v);
      C[(size_t)(m0 + (half << 3) + r) * ldc + nB] = v;
    }
  }
}

// ---------------------------------------------------------------------------
// Degree / scatter helpers
// ---------------------------------------------------------------------------
__global__ void k_fill1(float* __restrict__ p, int n) {
  int i = blockIdx.x * blockDim.x + threadIdx.x;
  if (i < n) p[i] = 1.0f;               // self-loop contributes 1 to degree
}

__global__ void k_deg_edges(const int* __restrict__ dst, float* __restrict__ deg,
                            int n) {
  int i = blockIdx.x * blockDim.x + threadIdx.x;
  if (i < n) unsafeAtomicAdd(&deg[dst[i]], 1.0f);
}

__global__ void k_rsqrt(float* __restrict__ p, int n) {
  int i = blockIdx.x * blockDim.x + threadIdx.x;
  if (i < n) p[i] = __frsqrt_rn(p[i]);  // deg >= 1 always
}

// out[g,f] = dinv[g]^2 * m[g,f]   (self-loop term of the scatter)
__global__ void k_self_init(const float* __restrict__ m,
                            const float* __restrict__ dinv,
                            float* __restrict__ out, int n, int logF) {
  int i = blockIdx.x * blockDim.x + threadIdx.x;
  if (i < n) {
    int g = i >> logF;
    float dv = dinv[g];
    out[i] = dv * dv * m[i];
  }
}

// broadcast bias into z before split-K atomic accumulation
__global__ void k_bias_bcast(float* __restrict__ p, const float* __restrict__ b,
                             int n, int fmask) {
  int i = blockIdx.x * blockDim.x + threadIdx.x;
  if (i < n) p[i] = b[i & fmask];
}

// EPB edges per block: out[dst,:] += dinv[s]*dinv[d] * m[src,:]
// Prefetch the next edge's source row (global_prefetch_b8) while the current
// row's atomics are in flight; one prefetch per 128B cacheline.
template <int F, int EPB>
__global__ void k_scatter_edges(const int* __restrict__ src,
                                const int* __restrict__ dst,
                                const float* __restrict__ dinv,
                                const float* __restrict__ m,
                                float* __restrict__ out) {
  const int ebase = blockIdx.x * EPB;
  const int tid = threadIdx.x;
#pragma unroll
  for (int i = 0; i < EPB; ++i) {
    const int e = ebase + i;
    const int s = src[e];               // uniform across block -> SGPR
    const int d = dst[e];
    if (i + 1 < EPB) {
      const int sn = src[e + 1];
      if (tid < F / 32)                 // one lane per 128B line of next row
        __builtin_prefetch(m + (size_t)sn * F + tid * 32, 0, 0);
    }
    const float norm = dinv[s] * dinv[d];
    const float* mrow = m + (size_t)s * F;
    float* orow = out + (size_t)d * F;
    for (int f = tid; f < F; f += (int)blockDim.x)
      unsafeAtomicAdd(&orow[f], norm * mrow[f]);
  }
}

// h[i] = (celu?)(h[i] + b[i & (F-1)])   in-place bias (+activation)
template <bool CELU>
__global__ void k_bias_act(float* __restrict__ h, const float* __restrict__ b,
                           int n, int fmask) {
  int i = blockIdx.x * blockDim.x + threadIdx.x;
  if (i < n) {
    float v = h[i] + b[i & fmask];
    if (CELU) v = celu1(v);
    h[i] = v;
  }
}

// ---------------------------------------------------------------------------
extern "C" void kernel_launch(void* const* d_in, const int* in_sizes, int n_in,
                              void* d_out, int out_size, void* d_ws,
                              size_t ws_size, hipStream_t stream) {
  (void)in_sizes; (void)n_in; (void)out_size; (void)ws_size;

  const float* x    = (const float*)d_in[0];   // [128, 20000]
  const int*   eidx = (const int*)d_in[1];     // [2, 640000] (jax x64-off -> int32)
  const float* W1   = (const float*)d_in[2];   // [128, 512]
  const float* b1   = (const float*)d_in[3];   // [512]
  const float* W2   = (const float*)d_in[4];   // [512, 128]
  const float* b2   = (const float*)d_in[5];   // [128]
  const float* Wdr  = (const float*)d_in[6];   // [20000, 64]
  const float* bdr  = (const float*)d_in[7];   // [64]
  const float* Wr1  = (const float*)d_in[8];   // [64, 256]
  const float* br1  = (const float*)d_in[9];   // [256]
  const float* Wr2  = (const float*)d_in[10];  // [256, 20000]
  const float* br2  = (const float*)d_in[11];  // [20000]

  const int* src = eidx;                // edge_index[0]
  const int* dst = eidx + NEDGES;       // edge_index[1]

  float* out  = (float*)d_out;
  float* z    = out;                    // [128, 64]
  float* y    = out + (size_t)NBATCH * TDIM;  // [128, 20000]

  // workspace: two big reusable regions + a1 + dinv  (~82 MB)
  float* ws    = (float*)d_ws;
  float* mbuf  = ws;                                   // m1 [20000,512] -> m2 [20000,128]
  float* hbuf  = ws + (size_t)NGENES * FMID;           // h1 [20000,512] -> h2 [20000,128]
  float* a1    = hbuf + (size_t)NGENES * FMID;         // [128, 256]
  float* dinv  = a1 + (size_t)NBATCH * RMID;           // [20000]

  const int T = 256;

  // --- degrees: deg = 1 + #incoming, dinv = rsqrt(deg) ----------------------
  k_fill1<<<(NGENES + T - 1) / T, T, 0, stream>>>(dinv, NGENES);
  k_deg_edges<<<(NEDGES + T - 1) / T, T, 0, stream>>>(dst, dinv, NEDGES);
  k_rsqrt<<<(NGENES + T - 1) / T, T, 0, stream>>>(dinv, NGENES);

  // --- layer 1: m1 = x^T @ W1  (A transposed access into x) -----------------
  {
    int tiles = (NGENES / 16) * (FMID / 16);           // 40000 waves
    wmma_gemm_f32<true, false, false, false, false>
        <<<(tiles + 7) / 8, 256, 0, stream>>>(
            x, W1, nullptr, mbuf, NGENES, FMID, NBATCH, NGENES, FMID, FMID, 1);
  }
  // scatter: h1 = dinv^2*m1 + sum_edges, then +b1, celu
  {
    int n = NGENES * FMID;
    k_self_init<<<(n + T - 1) / T, T, 0, stream>>>(mbuf, dinv, hbuf, n, 9);
    k_scatter_edges<FMID, 4><<<NEDGES / 4, 256, 0, stream>>>(src, dst, dinv,
                                                             mbuf, hbuf);
    k_bias_act<true><<<(n + T - 1) / T, T, 0, stream>>>(hbuf, b1, n, FMID - 1);
  }

  // --- layer 2: m2 = h1 @ W2 ------------------------------------------------
  {
    int tiles = (NGENES / 16) * (NBATCH / 16);         // 10000 waves
    wmma_gemm_f32<false, false, false, false, false>
        <<<(tiles + 7) / 8, 256, 0, stream>>>(
            hbuf, W2, nullptr, mbuf, NGENES, NBATCH, FMID, FMID, NBATCH, NBATCH, 1);
  }
  // scatter: h2 = dinv^2*m2 + sum_edges + b2 (no celu; celu fused into z GEMM)
  {
    int n = NGENES * NBATCH;
    k_self_init<<<(n + T - 1) / T, T, 0, stream>>>(mbuf, dinv, hbuf, n, 7);
    k_scatter_edges<NBATCH, 4><<<NEDGES / 4, 128, 0, stream>>>(src, dst, dinv,
                                                               mbuf, hbuf);
    k_bias_act<false><<<(n + T - 1) / T, T, 0, stream>>>(hbuf, b2, n, NBATCH - 1);
  }

  // --- z = celu(h2)^T @ W_dr + b_dr  ([128,20000]@[20000,64]) ---------------
  // split-K (20 splits of K=1000) with atomic reduction into bias-seeded z
  {
    const int nsplit = 20;
    int nz = NBATCH * TDIM;
    k_bias_bcast<<<(nz + T - 1) / T, T, 0, stream>>>(z, bdr, nz, TDIM - 1);
    int tiles = (NBATCH / 16) * (TDIM / 16);           // 32 tiles x 20 splits
    wmma_gemm_f32<true, true, false, false, true>
        <<<(tiles * nsplit + 7) / 8, 256, 0, stream>>>(
            hbuf, Wdr, nullptr, z, NBATCH, TDIM, NGENES, NBATCH, TDIM, TDIM,
            nsplit);
  }

  // --- a1 = celu(z @ W_r1 + b_r1)  ([128,64]@[64,256]) ----------------------
  {
    int tiles = (NBATCH / 16) * (RMID / 16);           // 128 waves
    wmma_gemm_f32<false, false, true, true, false>
        <<<(tiles + 7) / 8, 256, 0, stream>>>(
            z, Wr1, br1, a1, NBATCH, RMID, TDIM, TDIM, RMID, RMID, 1);
  }

  // --- y = celu(a1 @ W_r2 + b_r2)  ([128,256]@[256,20000]) ------------------
  {
    int tiles = (NBATCH / 16) * (NGENES / 16);         // 10000 waves
    wmma_gemm_f32<false, false, true, true, false>
        <<<(tiles + 7) / 8, 256, 0, stream>>>(
            a1, Wr2, br2, y, NBATCH, NGENES, RMID, RMID, NGENES, NGENES, 1);
  }
}